// RPFOModule_2602750181484
// MI455X (gfx1250) — compile-verified
//
#include <hip/hip_runtime.h>
#include <math.h>

// ---------------------------------------------------------------------------
// RPFO transform: out[b,j,p] = sum_k (Off[b,k,p]*4 + mean[k,p]) * M[b][j][k]
//                 + t[b][j]   (with /280 folded into M and t).
// HBM-bound (0.75 FLOP/byte): global loads via CDNA5 async->LDS (512B/instr),
// global stores full-width float4. The 3x3(+t) affine runs on the matrix pipe
// as V_WMMA_F32_16X16X4_F32 (K=4 carries translation), re-layouted through
// LDS so WMMA costs zero HBM width.
// ---------------------------------------------------------------------------

constexpr int   TPB      = 256;        // threads per block (8 waves, wave32)
constexpr int   PPT      = 4;          // pixels per thread (float4)
constexpr int   TILE_PIX = TPB * PPT;  // 1024 pixels per tile
constexpr int   TILES    = 4;          // tiles per block
constexpr int   PLANE    = 65536;      // 256*256 pixels per channel plane
constexpr int   PL       = TILE_PIX + 16; // padded LDS plane stride (floats)
constexpr float PI_F     = 3.14159265358979323846f;

typedef int   v4i __attribute__((ext_vector_type(4)));
typedef float v2f __attribute__((ext_vector_type(2)));
typedef float v8f __attribute__((ext_vector_type(8)));
using gv4i_p = v4i __attribute__((address_space(1)))*;   // global
using lv4i_p = v4i __attribute__((address_space(3)))*;   // LDS

#define HAVE_WMMA4 __has_builtin(__builtin_amdgcn_wmma_f32_16x16x4_f32)

// ---- CDNA5 async global->LDS 16B copy (per-lane), tracked by ASYNCcnt ----
__device__ __forceinline__ void async_copy16(const float* g, float* l) {
  unsigned long long gaddr  = (unsigned long long)g;
  unsigned           ldsoff = (unsigned)(unsigned long long)l; // low 32b = LDS offset
#if __has_builtin(__builtin_amdgcn_global_load_async_to_lds_b128)
  __builtin_amdgcn_global_load_async_to_lds_b128(
      (gv4i_p)gaddr, (lv4i_p)ldsoff, /*offset=*/0, /*cpol=*/0);
#else
  asm volatile("global_load_async_to_lds_b128 %0, %1, off"
               :: "v"(ldsoff), "v"(gaddr) : "memory");
#endif
}

template <int N> __device__ __forceinline__ void wait_asynccnt() {
#if __has_builtin(__builtin_amdgcn_s_wait_asynccnt)
  __builtin_amdgcn_s_wait_asynccnt(N);
#else
  asm volatile("s_wait_asynccnt %0" :: "n"(N) : "memory");
#endif
}

template <int N> __device__ __forceinline__ void wait_dscnt_n() {
#if __has_builtin(__builtin_amdgcn_s_wait_dscnt)
  __builtin_amdgcn_s_wait_dscnt(N);
#else
  asm volatile("s_wait_dscnt %0" :: "n"(N) : "memory");
#endif
}

// ---- float4 helpers -------------------------------------------------------
__device__ __forceinline__ float4 f4fma(float4 a, float s, float4 b) {
  return make_float4(fmaf(a.x, s, b.x), fmaf(a.y, s, b.y),
                     fmaf(a.z, s, b.z), fmaf(a.w, s, b.w));
}
__device__ __forceinline__ float4 f4bcast(float v) { return make_float4(v, v, v, v); }

// ---------------------------------------------------------------------------
// Kernel 1: per-batch constants. 12 floats per batch into d_ws:
//   [m00 m01 m02 t0 | m10 m11 m12 t1 | m20 m21 m22 t2]
// M[j][k] = r[j][k] * s_k / 280,  s_k = 20*S*(1,-1,1)_k,  t_j = T_j*300/280
// ---------------------------------------------------------------------------
__global__ void rpfo_prep_kernel(const float* __restrict__ R,
                                 const float* __restrict__ T,
                                 const float* __restrict__ S,
                                 float* __restrict__ C, int B) {
  int b = blockIdx.x * blockDim.x + threadIdx.x;
  if (b >= B) return;

  float ax = R[b * 3 + 0] * PI_F;
  float ay = R[b * 3 + 1] * PI_F;
  float az = R[b * 3 + 2] * PI_F;
  float sx, cx, sy, cy, sz, cz;
  sincosf(ax, &sx, &cx);
  sincosf(ay, &sy, &cy);
  sincosf(az, &sz, &cz);

  float r[3][3];
  r[0][0] = cy * cz;                r[0][1] = cy * sz;                r[0][2] = -sy;
  r[1][0] = sx * sy * cz - cx * sz; r[1][1] = sx * sy * sz + cx * cz; r[1][2] = sx * cy;
  r[2][0] = cx * sy * cz + sx * sz; r[2][1] = cx * sy * sz - sx * cz; r[2][2] = cx * cy;

  const float sscale  = S[b] * (10000.0f / 500.0f);   // 20*S
  const float inv280  = 1.0f / 280.0f;
  const float sk[3]   = {sscale * inv280, -sscale * inv280, sscale * inv280};
  const float t_scale = 300.0f * inv280;

  float* cb = C + b * 12;
#pragma unroll
  for (int j = 0; j < 3; ++j) {
#pragma unroll
    for (int k = 0; k < 3; ++k) cb[j * 4 + k] = r[j][k] * sk[k];
    cb[j * 4 + 3] = T[b * 3 + j] * t_scale;
  }
}

// ---------------------------------------------------------------------------
// Kernel 2: async->LDS double-buffered stream + WMMA affine transform.
// grid = (PLANE/(TILE_PIX*TILES), B), block = 256 (8 waves).
// All DS ordering is wave-internal (DS ops are in-order per wave): no barriers,
// only s_wait_asynccnt / s_wait_dscnt.
// ---------------------------------------------------------------------------
__global__ void __launch_bounds__(TPB)
rpfo_transform_kernel(const float* __restrict__ Off,
                      const float* __restrict__ Mean,
                      const float* __restrict__ C,
                      float* __restrict__ Out) {
  __shared__ float lds[2][6][PL];   // staged {off_x/y/z, mean_x/y/z} x2 buffers
  __shared__ float outl[3][PL];     // WMMA output restage (wave-private ranges)

  const int b    = blockIdx.y;
  const int tid  = threadIdx.x;
  const int q    = tid * PPT;                       // pixel offset in tile
  const int base = blockIdx.x * (TILE_PIX * TILES); // first pixel of block

  const float* offb = Off + (size_t)b * 3 * PLANE;
  float*       outb = Out + (size_t)b * 3 * PLANE;

  const float* cb = C + b * 12;                     // 16B-aligned (48B stride)
  const float4 c0 = *(const float4*)(cb + 0);       // {m00,m01,m02,t0}
  const float4 c1 = *(const float4*)(cb + 4);
  const float4 c2 = *(const float4*)(cb + 8);

  // Wave/lane geometry for the WMMA layouts
  const int wave  = tid >> 5;        // 0..7 : owns tile pixels [wave*128, +128)
  const int lane  = tid & 31;
  const int half  = lane >> 4;       // 0 = lanes 0-15, 1 = lanes 16-31
  const int ln    = lane & 15;

#if HAVE_WMMA4
  // A-matrix 16x4: lanes 0-15 hold K=0 (VGPR0) / K=1 (VGPR1); lanes 16-31 hold
  // K=2 / K=3. Rows: A[j][0..2]=M[j][*], A[j][3]=t_j (j<3), rows 3-15 zero.
  const float4 cr = (ln == 1) ? c1 : ((ln == 2) ? c2 : c0);
  const float  a0 = (ln < 3) ? (half ? cr.z : cr.x) : 0.0f;  // K=0 / K=2
  const float  a1 = (ln < 3) ? (half ? cr.w : cr.y) : 0.0f;  // K=1 / K=3
  const v2f    av = {a0, a1};
#endif

  auto issue = [&](int t) {                         // 6 async b128 per wave
    float*    Lb  = &lds[t & 1][0][0];
    const int pix = base + t * TILE_PIX + q;
#pragma unroll
    for (int c = 0; c < 3; ++c)
      async_copy16(offb + (size_t)c * PLANE + pix, Lb + c * PL + q);
#pragma unroll
    for (int c = 0; c < 3; ++c)
      async_copy16(Mean + (size_t)c * PLANE + pix, Lb + (3 + c) * PL + q);
  };

  auto compute = [&](int t) {
    const float* Lb = &lds[t & 1][0][0];
#if HAVE_WMMA4
    // ---- matrix-pipe path: 8 WMMAs of 16 pixels per wave ----
#pragma unroll
    for (int it = 0; it < 8; ++it) {
      const int p0 = wave * 128 + it * 16;          // tile-local pixel of N=0
      const int p  = p0 + ln;
      // B-matrix 4x16: VGPR0 = rows K=0 (lanes 0-15) / K=2 (lanes 16-31),
      //                VGPR1 = rows K=1 / K=3 (row 3 == constant 1.0)
      const int ch0 = half ? 2 : 0;
      const float b0 = fmaf(Lb[ch0 * PL + p], 4.0f, Lb[(3 + ch0) * PL + p]);
      const float py = fmaf(Lb[1   * PL + p], 4.0f, Lb[4        * PL + p]);
      const float b1 = half ? 1.0f : py;
      const v2f bv = {b0, b1};
      v8f acc = {};
      acc = __builtin_amdgcn_wmma_f32_16x16x4_f32(
          false, av, false, bv, (short)0, acc, false, false);
      // D: channel j = VGPR j, lanes 0-15 (N = lane). Restage to LDS planes.
      if (lane < 16) {
        outl[0][p0 + lane] = acc[0];
        outl[1][p0 + lane] = acc[1];
        outl[2][p0 + lane] = acc[2];
      }
    }
    // Readback own wave's pixels (DS in-order per wave) + full-width stores
    const int pix = base + t * TILE_PIX + q;
    float4 o0 = *(const float4*)(&outl[0][0] + q);
    float4 o1 = *(const float4*)(&outl[1][0] + q);
    float4 o2 = *(const float4*)(&outl[2][0] + q);
    *(float4*)(outb + (size_t)0 * PLANE + pix) = o0;
    *(float4*)(outb + (size_t)1 * PLANE + pix) = o1;
    *(float4*)(outb + (size_t)2 * PLANE + pix) = o2;
#else
    // ---- VALU fallback: float4 streaming transform ----
    const int pix = base + t * TILE_PIX + q;
    float4 ox = *(const float4*)(Lb + 0 * PL + q);
    float4 oy = *(const float4*)(Lb + 1 * PL + q);
    float4 oz = *(const float4*)(Lb + 2 * PL + q);
    float4 mx = *(const float4*)(Lb + 3 * PL + q);
    float4 my = *(const float4*)(Lb + 4 * PL + q);
    float4 mz = *(const float4*)(Lb + 5 * PL + q);
    float4 px = f4fma(ox, 4.0f, mx);
    float4 py = f4fma(oy, 4.0f, my);
    float4 pz = f4fma(oz, 4.0f, mz);
    float4 o0 = f4fma(pz, c0.z, f4fma(py, c0.y, f4fma(px, c0.x, f4bcast(c0.w))));
    float4 o1 = f4fma(pz, c1.z, f4fma(py, c1.y, f4fma(px, c1.x, f4bcast(c1.w))));
    float4 o2 = f4fma(pz, c2.z, f4fma(py, c2.y, f4fma(px, c2.x, f4bcast(c2.w))));
    *(float4*)(outb + (size_t)0 * PLANE + pix) = o0;
    *(float4*)(outb + (size_t)1 * PLANE + pix) = o1;
    *(float4*)(outb + (size_t)2 * PLANE + pix) = o2;
#endif
  };

  // Double-buffered pipeline (manually unrolled: compile-time wait immediates)
  issue(0);
  issue(1);
  wait_asynccnt<6>();  compute(0);  wait_dscnt_n<0>();  issue(2);
  wait_asynccnt<6>();  compute(1);  wait_dscnt_n<0>();  issue(3);
  wait_asynccnt<6>();  compute(2);
  wait_asynccnt<0>();  compute(3);
}

// ---------------------------------------------------------------------------
extern "C" void kernel_launch(void* const* d_in, const int* in_sizes, int n_in,
                              void* d_out, int out_size, void* d_ws, size_t ws_size,
                              hipStream_t stream) {
  const float* Off  = (const float*)d_in[0];   // (B,3,256,256)
  const float* R    = (const float*)d_in[1];   // (B,3)
  const float* T    = (const float*)d_in[2];   // (B,1,3)
  const float* S    = (const float*)d_in[3];   // (B,1)
  const float* Mean = (const float*)d_in[4];   // (3,256,256)
  float*       Out  = (float*)d_out;           // (B,3,256,256)

  const int B = in_sizes[3];                   // 256
  float* consts = (float*)d_ws;                // B*12 floats

  rpfo_prep_kernel<<<(B + 255) / 256, 256, 0, stream>>>(R, T, S, consts, B);

  dim3 grid(PLANE / (TILE_PIX * TILES), B);    // (16, 256)
  rpfo_transform_kernel<<<grid, TPB, 0, stream>>>(Off, Mean, consts, Out);
}